// RoIAlign_1726576857320
// MI455X (gfx1250) — compile-verified
//
#include <hip/hip_runtime.h>
#include <stdint.h>

// Problem constants (fixed by the reference setup).
#define CROP    7
#define NBOX    512
#define CH      256
#define FH      200
#define FW      304
#define PPB     (CROP * CROP)          // 49 sample positions per box
#define PERBOX  (CH * PPB)             // 12544 outputs per box

__global__ __launch_bounds__(256) void roialign_crop_resize_kernel(
    const float* __restrict__ fm,      // (2, 256, 200, 304)
    const float* __restrict__ boxes,   // (512, 4) as x1,y1,x2,y2
    const int*   __restrict__ box_ind, // (512,)
    float*       __restrict__ out)     // (512, 256, 7, 7)
{
    // Per-box uniform parameters staged through LDS by the CDNA5 async
    // copy engine (GLOBAL_LOAD_ASYNC_TO_LDS, tracked by ASYNCcnt).
    __shared__ __align__(16) float sprm[8]; // [0..3]=box, [4]=box_ind bits

    const int gid = blockIdx.x;                          // 0 .. NBOX*49-1
    const int m   = gid / PPB;                           // box id (block-uniform)
    const int idx = (gid - m * PPB) * 256 + threadIdx.x; // 0 .. 12543 within box

    if (threadIdx.x == 0) {
        const uint64_t gbox = (uint64_t)(uintptr_t)(boxes + 4 * m);
        const uint64_t gbi  = (uint64_t)(uintptr_t)(box_ind + m);
        // Generic->LDS address: low 32 bits of the flat pointer ARE the LDS
        // byte offset (aperture rule).
        const uint32_t lbox = (uint32_t)(uintptr_t)(&sprm[0]);
        const uint32_t lbi  = (uint32_t)(uintptr_t)(&sprm[4]);
        asm volatile(
            "global_load_async_to_lds_b128 %0, %1, off\n\t"
            "global_load_async_to_lds_b32  %2, %3, off\n\t"
            "s_wait_asynccnt 0x0"
            :
            : "v"(lbox), "v"(gbox), "v"(lbi), "v"(gbi)
            : "memory");
    }
    __syncthreads();

    // Decode (c, iy, ix) from the flat per-box output index so that
    // consecutive lanes write consecutive addresses (coalesced stores).
    const int c   = idx / PPB;
    const int pos = idx - c * PPB;
    const int iy  = pos / CROP;
    const int ix  = pos - iy * CROP;

    const float x1 = sprm[0];
    const float y1 = sprm[1];
    const float x2 = sprm[2];
    const float y2 = sprm[3];
    const int   b  = __float_as_int(sprm[4]);   // raw bits: untouched by load

    // Reference math reduces to: in = p1 + spacing/2 - 0.5 + i*spacing
    const float sw  = (x2 - x1) * (1.0f / (float)CROP);
    const float sh  = (y2 - y1) * (1.0f / (float)CROP);
    const float in_x = x1 + 0.5f * sw - 0.5f + (float)ix * sw;
    const float in_y = y1 + 0.5f * sh - 0.5f + (float)iy * sh;

    const bool valid = (in_x >= 0.0f) && (in_x <= (float)(FW - 1)) &&
                       (in_y >= 0.0f) && (in_y <= (float)(FH - 1));

    // Lerp fractions from UNCLAMPED floor (matches reference exactly).
    const float lf = floorf(in_x);
    const float tf = floorf(in_y);
    const float lx = in_x - lf;
    const float ly = in_y - tf;

    // Clamped integer corner indices for addressing.
    const int li = min(max((int)lf, 0), FW - 1);
    const int ri = min(max((int)ceilf(in_x), 0), FW - 1);
    const int ti = min(max((int)tf, 0), FH - 1);
    const int bi = min(max((int)ceilf(in_y), 0), FH - 1);

    const size_t plane = (size_t)(b * CH + c) * (size_t)(FH * FW);
    const float* rowT = fm + plane + (size_t)ti * FW;
    const float* rowB = fm + plane + (size_t)bi * FW;

    // 4 corner gathers (feature map stays RT-cached: it's re-read across
    // overlapping boxes and fits in the 192MB L2).
    const float tl = rowT[li];
    const float tr = rowT[ri];
    const float bl = rowB[li];
    const float br = rowB[ri];

    const float topv = tl + (tr - tl) * lx;
    const float botv = bl + (br - bl) * lx;
    float v = topv + (botv - topv) * ly;
    v = valid ? v : 0.0f;

    // Streamed, never re-read output: non-temporal store (TH=NT) so the
    // 25.7MB output stream does not evict feature-map lines from L2.
    __builtin_nontemporal_store(v, out + (size_t)m * PERBOX + (size_t)idx);
}

extern "C" void kernel_launch(void* const* d_in, const int* in_sizes, int n_in,
                              void* d_out, int out_size, void* d_ws, size_t ws_size,
                              hipStream_t stream) {
    const float* fm      = (const float*)d_in[0];
    const float* boxes   = (const float*)d_in[1];
    const int*   box_ind = (const int*)d_in[2];
    float*       out     = (float*)d_out;

    // One thread per output element: NBOX*49 blocks of 256 threads
    // (8 wave32 waves per block) = 6.42M threads.
    dim3 grid(NBOX * PPB);
    dim3 block(256);
    hipLaunchKernelGGL(roialign_crop_resize_kernel, grid, block, 0, stream,
                       fm, boxes, box_ind, out);
    (void)in_sizes; (void)n_in; (void)out_size; (void)d_ws; (void)ws_size;
}